// SchrodingerAttention_75239237091374
// MI455X (gfx1250) — compile-verified
//
#include <hip/hip_runtime.h>
#include <math.h>

#define BSZv   4
#define LSEQ   4096
#define DIMv   512
#define KBANDS 4
#define N2D    (2*DIMv)         // 1024
#define MROWS  (BSZv*LSEQ)      // 16384
#define CHUNK  512
#define NCHUNK (LSEQ/CHUNK)     // 8

typedef __attribute__((ext_vector_type(2))) float v2f;
typedef __attribute__((ext_vector_type(4))) float v4f;
typedef __attribute__((ext_vector_type(8))) float v8f;

#if defined(__has_builtin)
#if __has_builtin(__builtin_amdgcn_tensor_load_to_lds) && __has_builtin(__builtin_amdgcn_s_wait_tensorcnt)
#define HAVE_TDM 1
#endif
#endif

#ifdef HAVE_TDM
typedef __attribute__((ext_vector_type(4))) unsigned int u32x4;
typedef __attribute__((ext_vector_type(8))) int i32x8;
typedef __attribute__((ext_vector_type(4))) int i32x4;
#endif

// ---------------------------------------------------------------------------
// GEMM: P[m][n] = sum_k X[m][k] * W[n][k] + bias[n]
// X: MROWS x DIMv, W: N2D x DIMv, P: MROWS x N2D
// WMMA f32 16x16x4; block = 128 threads (4 waves); block tile = 16 x 256;
// each wave computes 16x64 (4 accumulators reusing one A-fragment).
// x-tile (16x512 f32, 32KB) staged into LDS by the Tensor Data Mover.
// ---------------------------------------------------------------------------
__global__ __launch_bounds__(128)
void gemm_wmma_f32(const float* __restrict__ X, const float* __restrict__ W,
                   const float* __restrict__ bias, float* __restrict__ P)
{
    __shared__ float xs[16 * DIMv];            // 32 KB x-tile
    const int m0  = blockIdx.x * 16;
    const int n0  = blockIdx.y * 256;
    const int tid = threadIdx.x;

#ifdef HAVE_TDM
    if (tid < 32) {                            // wave 0 issues the TDM descriptor
        unsigned int lds_off = (unsigned int)(size_t)(&xs[0]);
        unsigned long long ga = (unsigned long long)(const void*)(X + (size_t)m0 * DIMv);
        // D# group0: count=1 | lds_addr | global_addr[56:0] | type=2
        u32x4 g0 = { 1u, lds_off, (unsigned int)ga,
                     (unsigned int)((ga >> 32) & 0x1FFFFFFu) | (2u << 30) };
        // D# group1: data_size=4B; tensor_dim0=512; tensor_dim1=16384;
        //            tile_dim0=512; tile_dim1=16; tensor_dim0_stride=512
        i32x8 g1 = { (int)(2u << 16),
                     (int)(512u << 16),
                     (int)((MROWS & 0xFFFFu) << 16),
                     (int)(512u << 16),
                     16,
                     512,
                     0, 0 };
        i32x4 gz4 = { 0, 0, 0, 0 };
        i32x8 gz8 = { 0, 0, 0, 0, 0, 0, 0, 0 };
        __builtin_amdgcn_tensor_load_to_lds(g0, g1, gz4, gz4, gz8, 0);
        __builtin_amdgcn_s_wait_tensorcnt(0);
    }
#else
    {
        const v4f* src = (const v4f*)(X + (size_t)m0 * DIMv);
        v4f* dst = (v4f*)xs;
        #pragma unroll
        for (int i = 0; i < 16; ++i)
            dst[tid + i * 128] = src[tid + i * 128];
    }
#endif
    __syncthreads();

    const int wave = tid >> 5;
    const int lane = tid & 31;
    const int half = lane >> 4;     // K-halves: lanes 0-15 hold K,K+1; 16-31 hold K+2,K+3
    const int l16  = lane & 15;
    const int ncol = n0 + wave * 64;

    v8f c0 = {}; v8f c1 = {}; v8f c2 = {}; v8f c3 = {};
    const float* wrow0 = W + (size_t)(ncol +  0 + l16) * DIMv + half * 2;
    const float* wrow1 = W + (size_t)(ncol + 16 + l16) * DIMv + half * 2;
    const float* wrow2 = W + (size_t)(ncol + 32 + l16) * DIMv + half * 2;
    const float* wrow3 = W + (size_t)(ncol + 48 + l16) * DIMv + half * 2;
    const float* arow  = xs + l16 * DIMv + half * 2;

    #pragma unroll 4
    for (int k = 0; k < DIMv; k += 4) {
        v2f a  = *(const v2f*)(arow + k);       // ds_load_b64
        v2f b0 = *(const v2f*)(wrow0 + k);      // global b64 (L2-resident W)
        v2f b1 = *(const v2f*)(wrow1 + k);
        v2f b2 = *(const v2f*)(wrow2 + k);
        v2f b3 = *(const v2f*)(wrow3 + k);
        c0 = __builtin_amdgcn_wmma_f32_16x16x4_f32(false, a, false, b0,
                                                   (short)0, c0, false, false);
        c1 = __builtin_amdgcn_wmma_f32_16x16x4_f32(false, a, false, b1,
                                                   (short)0, c1, false, false);
        c2 = __builtin_amdgcn_wmma_f32_16x16x4_f32(false, a, false, b2,
                                                   (short)0, c2, false, false);
        c3 = __builtin_amdgcn_wmma_f32_16x16x4_f32(false, a, false, b3,
                                                   (short)0, c3, false, false);
    }

    const int col0 = ncol +  0 + l16;
    const int col1 = ncol + 16 + l16;
    const int col2 = ncol + 32 + l16;
    const int col3 = ncol + 48 + l16;
    const float bi0 = bias[col0];
    const float bi1 = bias[col1];
    const float bi2 = bias[col2];
    const float bi3 = bias[col3];
    #pragma unroll
    for (int r = 0; r < 8; ++r) {
        int row = m0 + r + half * 8;            // C/D layout: lanes>=16 hold M=r+8
        float* pr = P + (size_t)row * N2D;
        pr[col0] = c0[r] + bi0;
        pr[col1] = c1[r] + bi1;
        pr[col2] = c2[r] + bi2;
        pr[col3] = c3[r] + bi3;
    }
}

// ---------------------------------------------------------------------------
// gate[row] = sigmoid(x.Wg + bg) * (1 + tanh(sgain)*mean|x - z|)
// ---------------------------------------------------------------------------
__global__ __launch_bounds__(128)
void gate_kernel(const float* __restrict__ X, const float* __restrict__ Z,
                 const float* __restrict__ Wg, const float* __restrict__ bg,
                 const float* __restrict__ sgain, float* __restrict__ gate)
{
    __shared__ float red[256];
    const int row = blockIdx.x;
    const int tid = threadIdx.x;
    const float* xr = X + (size_t)row * DIMv;
    const float* zr = Z + (size_t)row * DIMv;
    float dot = 0.f, sd = 0.f;
    #pragma unroll
    for (int d = tid; d < DIMv; d += 128) {
        float xv = xr[d];
        dot += xv * Wg[d];
        sd  += fabsf(xv - zr[d]);
    }
    red[tid] = dot; red[128 + tid] = sd;
    __syncthreads();
    for (int s = 64; s > 0; s >>= 1) {
        if (tid < s) { red[tid] += red[tid + s]; red[128 + tid] += red[128 + tid + s]; }
        __syncthreads();
    }
    if (tid == 0) {
        float g = 1.f / (1.f + expf(-(red[0] + bg[0])));
        float surprise = red[128] * (1.f / (float)DIMv);
        gate[row] = g * (1.f + tanhf(sgain[0]) * surprise);
    }
}

// ---------------------------------------------------------------------------
// Fused U-formation + chunk-local scan:
//   U[t,d] = gate[row]*Bv[d] * psi[row,d];  H[t] = a*H[t-1] + U[t]
//   a = exp(lam*|dt|), lam = -exp(log_gamma) + i*omega.
// Writes H (local scans) and per-chunk end values E.
// ---------------------------------------------------------------------------
__global__ void scan_local(const float* __restrict__ psi_raw,
                           const float* __restrict__ gate,
                           const float* __restrict__ Bv,
                           float* __restrict__ Hr, float* __restrict__ Hi,
                           float* __restrict__ E,
                           const float* __restrict__ log_gamma,
                           const float* __restrict__ omega,
                           const float* __restrict__ dtp)
{
    int tid = blockIdx.x * blockDim.x + threadIdx.x;   // 0 .. B*D*NCHUNK-1
    if (tid >= BSZv * DIMv * NCHUNK) return;
    int g = tid % (BSZv * DIMv);     // b*D + d  (consecutive threads -> consecutive d)
    int c = tid / (BSZv * DIMv);
    int d = g % DIMv;
    int b = g / DIMv;
    float dta = fabsf(dtp[0]);
    float dec = expf(-expf(log_gamma[d]) * dta);
    float ang = omega[d] * dta;
    float ar = dec * cosf(ang);
    float ai = dec * sinf(ang);
    float bv = Bv[d];
    size_t row0 = (size_t)b * LSEQ + (size_t)c * CHUNK;
    float hr = 0.f, hi = 0.f;
    for (int t = 0; t < CHUNK; ++t) {
        size_t row = row0 + t;
        float s  = gate[row] * bv;
        float ur = s * psi_raw[row * N2D + d];
        float ui = s * psi_raw[row * N2D + DIMv + d];
        float nr = ar * hr - ai * hi + ur;
        float ni = ar * hi + ai * hr + ui;
        hr = nr; hi = ni;
        Hr[row * DIMv + d] = hr;
        Hi[row * DIMv + d] = hi;
    }
    E[(size_t)(g * NCHUNK + c) * 2]     = hr;
    E[(size_t)(g * NCHUNK + c) * 2 + 1] = hi;
}

__global__ void scan_carry(const float* __restrict__ E, float* __restrict__ Cb,
                           const float* __restrict__ log_gamma,
                           const float* __restrict__ omega,
                           const float* __restrict__ dtp)
{
    int g = blockIdx.x * blockDim.x + threadIdx.x;
    if (g >= BSZv * DIMv) return;
    int d = g % DIMv;
    float dta = fabsf(dtp[0]);
    float n = (float)CHUNK;
    float dec = expf(-expf(log_gamma[d]) * dta * n);
    float ang = omega[d] * dta * n;
    float Ar = dec * cosf(ang);
    float Ai = dec * sinf(ang);
    float cr = 0.f, ci = 0.f;
    for (int c = 0; c < NCHUNK; ++c) {
        Cb[(size_t)(g * NCHUNK + c) * 2]     = cr;
        Cb[(size_t)(g * NCHUNK + c) * 2 + 1] = ci;
        float er = E[(size_t)(g * NCHUNK + c) * 2];
        float ei = E[(size_t)(g * NCHUNK + c) * 2 + 1];
        float nr = Ar * cr - Ai * ci + er;
        float ni = Ar * ci + Ai * cr + ei;
        cr = nr; ci = ni;
    }
}

// ---------------------------------------------------------------------------
// Fused: carry-apply (global scan fixup) + complex layer-norm over D +
// band gather + alpha dots vs phi + softmax over K + density + scatter.
// One block per (b,t) row.
// ---------------------------------------------------------------------------
__global__ __launch_bounds__(256)
void bands_kernel(const float* __restrict__ Hr, const float* __restrict__ Hi,
                  const float* __restrict__ Cb,
                  const float* __restrict__ phi_raw,
                  const int* __restrict__ band_idx,
                  const float* __restrict__ log_gamma,
                  const float* __restrict__ omega,
                  const float* __restrict__ dtp,
                  const float* __restrict__ taup, const float* __restrict__ betap,
                  float* __restrict__ out)
{
    __shared__ float hr[DIMv], hi[DIMv];
    __shared__ v4f   red[256];
    __shared__ float aR[KBANDS], aI[KBANDS], dK[KBANDS];
    const int row = blockIdx.x;
    const int tid = threadIdx.x;
    const size_t base = (size_t)row * DIMv;
    const int t  = row & (LSEQ - 1);
    const int b  = row >> 12;           // / LSEQ
    const int c  = t >> 9;              // / CHUNK
    const int tl = t & (CHUNK - 1);
    const float dta = fabsf(dtp[0]);
    const float pn  = (float)(tl + 1);

    float h0r = Hr[base + tid], h1r = Hr[base + tid + 256];
    float h0i = Hi[base + tid], h1i = Hi[base + tid + 256];
    if (c > 0) {                         // apply inter-chunk carry: H += carry * a^(tl+1)
        int g0 = b * DIMv + tid;
        int g1 = g0 + 256;
        float cr0 = Cb[(size_t)(g0 * NCHUNK + c) * 2];
        float ci0 = Cb[(size_t)(g0 * NCHUNK + c) * 2 + 1];
        float cr1 = Cb[(size_t)(g1 * NCHUNK + c) * 2];
        float ci1 = Cb[(size_t)(g1 * NCHUNK + c) * 2 + 1];
        float de0 = expf(-expf(log_gamma[tid]) * dta * pn);
        float de1 = expf(-expf(log_gamma[tid + 256]) * dta * pn);
        float an0 = omega[tid] * dta * pn;
        float an1 = omega[tid + 256] * dta * pn;
        float pr0 = de0 * cosf(an0), pi0 = de0 * sinf(an0);
        float pr1 = de1 * cosf(an1), pi1 = de1 * sinf(an1);
        h0r += cr0 * pr0 - ci0 * pi0;  h0i += cr0 * pi0 + ci0 * pr0;
        h1r += cr1 * pr1 - ci1 * pi1;  h1i += cr1 * pi1 + ci1 * pr1;
    }
    hr[tid] = h0r; hr[tid + 256] = h1r;
    hi[tid] = h0i; hi[tid + 256] = h1i;
    v4f acc = { h0r + h1r, h0r * h0r + h1r * h1r,
                h0i + h1i, h0i * h0i + h1i * h1i };
    red[tid] = acc;
    __syncthreads();
    for (int s = 128; s > 0; s >>= 1) {
        if (tid < s) red[tid] += red[tid + s];
        __syncthreads();
    }
    v4f tot = red[0];
    float mr = tot.x * (1.f / DIMv);
    float vr = tot.y * (1.f / DIMv) - mr * mr;
    float mi = tot.z * (1.f / DIMv);
    float vi = tot.w * (1.f / DIMv) - mi * mi;
    float ivr = 1.f / (sqrtf(vr) + 1e-6f);
    float ivi = 1.f / (sqrtf(vi) + 1e-6f);
    hr[tid]       = (hr[tid]       - mr) * ivr;
    hr[tid + 256] = (hr[tid + 256] - mr) * ivr;
    hi[tid]       = (hi[tid]       - mi) * ivi;
    hi[tid + 256] = (hi[tid + 256] - mi) * ivi;
    __syncthreads();

    const int wave = tid >> 5, lane = tid & 31;
    if (wave < KBANDS) {
        float ar = 0.f, ai = 0.f, dn = 0.f;
        #pragma unroll
        for (int j = lane; j < DIMv / KBANDS; j += 32) {
            int e = wave * (DIMv / KBANDS) + j;
            int d = band_idx[e];
            float hrn = hr[d], hin = hi[d];
            float pr = phi_raw[base * 2 + d];
            float pi = phi_raw[base * 2 + DIMv + d];
            ar += hrn * pr + hin * pi;      // conj(H)*P real
            ai += hrn * pi - hin * pr;      // conj(H)*P imag
            dn += hrn * hrn + hin * hin;
        }
        #pragma unroll
        for (int off = 16; off > 0; off >>= 1) {
            ar += __shfl_down(ar, off, 32);
            ai += __shfl_down(ai, off, 32);
            dn += __shfl_down(dn, off, 32);
        }
        if (lane == 0) {
            float inv = 1.f / (float)(DIMv / KBANDS);
            aR[wave] = ar * inv; aI[wave] = ai * inv; dK[wave] = dn * inv;
        }
    }
    __syncthreads();

    float tau = taup[0]; tau = tau < 1e-4f ? 1e-4f : tau;
    float beta = betap[0];
    float lg[KBANDS], mx = -1e30f;
    #pragma unroll
    for (int k = 0; k < KBANDS; ++k) {
        lg[k] = sqrtf(aR[k] * aR[k] + aI[k] * aI[k]) / tau;
        mx = fmaxf(mx, lg[k]);
    }
    float sum = 0.f;
    #pragma unroll
    for (int k = 0; k < KBANDS; ++k) { lg[k] = expf(lg[k] - mx); sum += lg[k]; }
    float afr[KBANDS], afi[KBANDS];
    #pragma unroll
    for (int k = 0; k < KBANDS; ++k) {
        float ck = lg[k] / sum;
        afr[k] = aR[k] * ck * (float)KBANDS + beta * dK[k];  // beta*density -> real part
        afi[k] = aI[k] * ck * (float)KBANDS;
    }
    #pragma unroll
    for (int e2 = 0; e2 < 2; ++e2) {
        int e = tid + e2 * 256;
        int k = e >> 7;                     // 128 elements per band
        int d = band_idx[e];
        out[base + d] = hr[d] * afr[k] - hi[d] * afi[k];
    }
}

// ---------------------------------------------------------------------------
extern "C" void kernel_launch(void* const* d_in, const int* in_sizes, int n_in,
                              void* d_out, int out_size, void* d_ws, size_t ws_size,
                              hipStream_t stream) {
    const float* x        = (const float*)d_in[0];
    const float* z_prev   = (const float*)d_in[1];
    const float* W_psi    = (const float*)d_in[2];
    const float* b_psi    = (const float*)d_in[3];
    const float* W_phi    = (const float*)d_in[4];
    const float* b_phi    = (const float*)d_in[5];
    const float* W_gate   = (const float*)d_in[6];
    const float* b_gate   = (const float*)d_in[7];
    const float* omega    = (const float*)d_in[8];
    const float* log_gam  = (const float*)d_in[9];
    const float* dt       = (const float*)d_in[10];
    const float* sgain    = (const float*)d_in[11];
    const float* tau      = (const float*)d_in[12];
    const float* beta     = (const float*)d_in[13];
    const float* B_vec    = (const float*)d_in[14];
    const int*   band_idx = (const int*)d_in[15];
    float* out = (float*)d_out;

    float* ws = (float*)d_ws;
    const size_t PSI  = 0;                                   // MROWS*N2D
    const size_t PHI  = PSI + (size_t)MROWS * N2D;           // MROWS*N2D
    const size_t HR   = PHI + (size_t)MROWS * N2D;           // MROWS*DIMv
    const size_t HI   = HR  + (size_t)MROWS * DIMv;          // MROWS*DIMv
    const size_t GATE = HI  + (size_t)MROWS * DIMv;          // MROWS
    const size_t EOFF = GATE + MROWS;                        // B*D*NC*2
    const size_t COFF = EOFF + (size_t)BSZv * DIMv * NCHUNK * 2;
    float* psi_raw = ws + PSI;
    float* phi_raw = ws + PHI;
    float* Hr      = ws + HR;
    float* Hi      = ws + HI;
    float* gate    = ws + GATE;
    float* Ebuf    = ws + EOFF;
    float* Cbuf    = ws + COFF;

    dim3 ggrid(MROWS / 16, N2D / 256);
    gemm_wmma_f32<<<ggrid, 128, 0, stream>>>(x, W_psi, b_psi, psi_raw);
    gemm_wmma_f32<<<ggrid, 128, 0, stream>>>(x, W_phi, b_phi, phi_raw);

    gate_kernel<<<MROWS, 128, 0, stream>>>(x, z_prev, W_gate, b_gate, sgain, gate);

    int nscan = BSZv * DIMv * NCHUNK;
    scan_local<<<(nscan + 255) / 256, 256, 0, stream>>>(psi_raw, gate, B_vec,
                                                        Hr, Hi, Ebuf, log_gam, omega, dt);
    scan_carry<<<(BSZv * DIMv + 255) / 256, 256, 0, stream>>>(Ebuf, Cbuf, log_gam, omega, dt);

    bands_kernel<<<MROWS, 256, 0, stream>>>(Hr, Hi, Cbuf, phi_raw, band_idx,
                                            log_gam, omega, dt, tau, beta, out);

    (void)in_sizes; (void)n_in; (void)out_size; (void)ws_size;
}